// IO_Layer_10694468567410
// MI455X (gfx1250) — compile-verified
//
#include <hip/hip_runtime.h>

typedef float v2f __attribute__((ext_vector_type(2)));
typedef float v4f __attribute__((ext_vector_type(4)));
typedef float v8f __attribute__((ext_vector_type(8)));

#define MEM_ROWS 131072
#define ESIZE 512
#define WAVES_PER_BLOCK 8
#define TPB 256
#define PARTIAL_STRIDE 1026   // A[512] | B[512] | Sw | Sr

// ---------------------------------------------------------------------------
// vec(1xK) @ W(Kx512) -> out(512), one wave per 16-column tile, using
// v_wmma_f32_16x16x4_f32. A-matrix: row 0 = vec, rows 1..15 = 0.
//   A layout (16x4 f32): lanes 0-15 = M, vgpr0/1 = K+0/K+1; lanes 16-31 = K+2/K+3
//   B layout (4x16 f32): lane n (0-15) vgpr0/1 = W[K+0][n], W[K+1][n];
//                        lane n+16     vgpr0/1 = W[K+2][n], W[K+3][n]
//   D row 0: acc vgpr0, lanes 0-15.
// ---------------------------------------------------------------------------
__global__ __launch_bounds__(32) void gemv_wmma(const float* __restrict__ vec,
                                                const float* __restrict__ W,
                                                float* __restrict__ out, int K) {
  const int lane = threadIdx.x;
  const int e0   = blockIdx.x * 16;
  const int half = lane >> 4;   // 0 or 1
  const int l15  = lane & 15;
  v8f acc = {0.f, 0.f, 0.f, 0.f, 0.f, 0.f, 0.f, 0.f};
  for (int k = 0; k < K; k += 4) {
    const int kk = k + half * 2;
    const float av0 = vec[kk];
    const float av1 = vec[kk + 1];
    v2f a, b;
    a.x = (l15 == 0) ? av0 : 0.0f;   // only M=0 row carries the vector
    a.y = (l15 == 0) ? av1 : 0.0f;
    b.x = W[(size_t)kk * ESIZE + e0 + l15];
    b.y = W[(size_t)(kk + 1) * ESIZE + e0 + l15];
    acc = __builtin_amdgcn_wmma_f32_16x16x4_f32(false, a, false, b,
                                                (short)0, acc, false, false);
  }
  if (lane < 16) out[e0 + lane] = acc[0];
}

// ---------------------------------------------------------------------------
// Single fused pass over memory (read once, 256 MB). Each wave owns whole
// rows; lane l covers elements [16l, 16l+16). Per row: d_w, d_r, n^2 via FMA,
// wave-wide shuffle reduce, then accumulate A += t_r*row, B += t_r*t_w*row in
// registers. Block combines 8 waves via LDS in a fixed order (deterministic),
// writes one partial record per block.
// ---------------------------------------------------------------------------
__global__ __launch_bounds__(TPB) void stream_pass(const float* __restrict__ mem,
                                                   const float* __restrict__ keys,
                                                   float* __restrict__ partial,
                                                   int nblocks) {
  __shared__ float sA[WAVES_PER_BLOCK][ESIZE];
  __shared__ float sB[WAVES_PER_BLOCK][ESIZE];
  __shared__ float sS[WAVES_PER_BLOCK][2];

  const int lane = threadIdx.x & 31;
  const int wave = threadIdx.x >> 5;
  const int gw   = blockIdx.x * WAVES_PER_BLOCK + wave;
  const int nw   = nblocks * WAVES_PER_BLOCK;
  const int eb   = lane * 16;

  float wk[16], rk[16], Ar[16], Br[16];
#pragma unroll
  for (int j = 0; j < 16; ++j) {
    wk[j] = keys[eb + j];
    rk[j] = keys[ESIZE + eb + j];
    Ar[j] = 0.f;
    Br[j] = 0.f;
  }
  float Sw = 0.f, Sr = 0.f;

  for (int m = gw; m < MEM_ROWS; m += nw) {
    const float* row = mem + (size_t)m * ESIZE + eb;
    int mn = m + nw; if (mn >= MEM_ROWS) mn = m;          // safe prefetch addr
    __builtin_prefetch(mem + (size_t)mn * ESIZE + eb, 0, 0);

    v4f v0 = __builtin_nontemporal_load((const v4f*)(row + 0));
    v4f v1 = __builtin_nontemporal_load((const v4f*)(row + 4));
    v4f v2 = __builtin_nontemporal_load((const v4f*)(row + 8));
    v4f v3 = __builtin_nontemporal_load((const v4f*)(row + 12));
    float v[16];
#pragma unroll
    for (int j = 0; j < 4; ++j) {
      v[j] = v0[j]; v[4 + j] = v1[j]; v[8 + j] = v2[j]; v[12 + j] = v3[j];
    }

    float dw = 0.f, dr = 0.f, n2 = 0.f;
#pragma unroll
    for (int j = 0; j < 16; ++j) {
      dw = fmaf(v[j], wk[j], dw);
      dr = fmaf(v[j], rk[j], dr);
      n2 = fmaf(v[j], v[j], n2);
    }
#pragma unroll
    for (int off = 16; off > 0; off >>= 1) {
      dw += __shfl_xor(dw, off);
      dr += __shfl_xor(dr, off);
      n2 += __shfl_xor(n2, off);
    }
    const float rn = 1.0f / sqrtf(n2);
    const float tw = dw * rn;   // sims_w * ||w_key||  (key norm cancels)
    const float tr = dr * rn;
    Sw += tw;
    Sr += tr;
    const float cb = tr * tw;
#pragma unroll
    for (int j = 0; j < 16; ++j) {
      Ar[j] = fmaf(tr, v[j], Ar[j]);
      Br[j] = fmaf(cb, v[j], Br[j]);
    }
  }

#pragma unroll
  for (int j = 0; j < 16; ++j) {
    sA[wave][eb + j] = Ar[j];
    sB[wave][eb + j] = Br[j];
  }
  if (lane == 0) { sS[wave][0] = Sw; sS[wave][1] = Sr; }
  __syncthreads();

  const int t = threadIdx.x;       // 256 threads -> elems t and t+256
  float a0 = 0.f, a1 = 0.f, b0 = 0.f, b1 = 0.f;
#pragma unroll
  for (int w = 0; w < WAVES_PER_BLOCK; ++w) {
    a0 += sA[w][t];       b0 += sB[w][t];
    a1 += sA[w][t + 256]; b1 += sB[w][t + 256];
  }
  float* p = partial + (size_t)blockIdx.x * PARTIAL_STRIDE;
  p[t] = a0; p[t + 256] = a1;
  p[512 + t] = b0; p[512 + t + 256] = b1;
  if (t == 0) { float s = 0.f; for (int w = 0; w < WAVES_PER_BLOCK; ++w) s += sS[w][0]; p[1024] = s; }
  if (t == 1) { float s = 0.f; for (int w = 0; w < WAVES_PER_BLOCK; ++w) s += sS[w][1]; p[1025] = s; }
}

// ---------------------------------------------------------------------------
// Reduce per-block partials (fixed order -> deterministic), compute
// r_vect = A/Sr - erase * B/(Sr*Sw), emit concat = [x0 | r_vect].
// ---------------------------------------------------------------------------
__global__ __launch_bounds__(512) void finalize(const float* __restrict__ partial,
                                                int G,
                                                const float* __restrict__ x,
                                                const float* __restrict__ erase,
                                                float* __restrict__ concat) {
  __shared__ float sS[2];
  const int e = threadIdx.x;   // 512 threads
  if (e < 2) {
    float s = 0.f;
    for (int g = 0; g < G; ++g) s += partial[(size_t)g * PARTIAL_STRIDE + 1024 + e];
    sS[e] = s;
  }
  __syncthreads();
  float A = 0.f, B = 0.f;
  for (int g = 0; g < G; ++g) {
    const float* p = partial + (size_t)g * PARTIAL_STRIDE;
    A += p[e];
    B += p[512 + e];
  }
  const float Sw = sS[0], Sr = sS[1];
  const float rv = A / Sr - erase[e] * (B / (Sr * Sw));
  concat[e] = x[e];
  concat[ESIZE + e] = rv;
}

// ---------------------------------------------------------------------------
extern "C" void kernel_launch(void* const* d_in, const int* in_sizes, int n_in,
                              void* d_out, int out_size, void* d_ws, size_t ws_size,
                              hipStream_t stream) {
  (void)in_sizes; (void)n_in; (void)out_size;
  const float* x    = (const float*)d_in[0];  // (1,512)
  const float* mem  = (const float*)d_in[1];  // (131072,512)
  const float* wkg  = (const float*)d_in[2];  // w_key_gen
  // d_in[3] = w_vect_gen: dead in the reference, never read
  const float* weg  = (const float*)d_in[4];  // w_erase_gen
  const float* rkg  = (const float*)d_in[5];  // r_key_gen
  const float* post = (const float*)d_in[6];  // (1024,512)
  float* out = (float*)d_out;

  float* ws      = (float*)d_ws;
  float* wkey    = ws;            // 512
  float* rkey    = ws + 512;      // 512 (keys buffer = [wkey|rkey], contiguous)
  float* erase   = ws + 1024;     // 512
  float* concat  = ws + 1536;     // 1024
  float* partial = ws + 4096;     // G * 1026

  int G = 256;
  const size_t avail = ws_size / sizeof(float);
  if (avail < (size_t)4096 + (size_t)G * PARTIAL_STRIDE) {
    long g = ((long)avail - 4096) / PARTIAL_STRIDE;
    G = (g < 1) ? 1 : (int)g;
  }

  // Key/erase generation: 1x512 @ 512x512, WMMA f32 16x16x4
  gemv_wmma<<<32, 32, 0, stream>>>(x, wkg, wkey, ESIZE);
  gemv_wmma<<<32, 32, 0, stream>>>(x, rkg, rkey, ESIZE);
  gemv_wmma<<<32, 32, 0, stream>>>(x, weg, erase, ESIZE);

  // Single fused 256 MB streaming pass over memory
  stream_pass<<<G, TPB, 0, stream>>>(mem, ws, partial, G);

  // Deterministic reduction + r_vect, build concat
  finalize<<<1, 512, 0, stream>>>(partial, G, x, erase, concat);

  // out = concat(1x1024) @ post(1024x512), WMMA f32 16x16x4
  gemv_wmma<<<32, 32, 0, stream>>>(concat, post, out, 2 * ESIZE);
}